// QueryBackbone_44770739093784
// MI455X (gfx1250) — compile-verified
//
#include <hip/hip_runtime.h>
#include <cmath>

typedef __attribute__((ext_vector_type(2))) float v2f;
typedef __attribute__((ext_vector_type(8))) float v8f;

#define BATCH 4
#define CH    256
#define NPTS  16384
#define ROWS  (BATCH*NPTS)   // 65536
#define HCH   64
#define NCLS  10
#define NQ    256
#define EPSF  1e-5f

__device__ __forceinline__ float gelu_tanh(float x){
  const float c = 0.7978845608028654f; // sqrt(2/pi)
  float x3 = x*x*x;
  return 0.5f*x*(1.0f + tanhf(c*(x + 0.044715f*x3)));
}

// ---------------- GEMM1: Y[r,j] = sum_c feats[b,c,n] * w1[c,j] ------------
// one wave -> 16 rows x 64 cols (4 N-tiles), K=256 via V_WMMA_F32_16X16X4_F32
__global__ void k_gemm1(const float* __restrict__ feats, const float* __restrict__ w1,
                        float* __restrict__ Y){
  int wid  = (blockIdx.x*blockDim.x + threadIdx.x) >> 5;
  int lane = threadIdx.x & 31;
  int hl   = lane >> 4;          // lane half (K split)
  int l15  = lane & 15;
  int r0 = wid * 16;
  int b  = r0 / NPTS;
  int n0 = r0 % NPTS;
  const float* fb = feats + (size_t)b*CH*NPTS;
  v8f acc0 = {}, acc1 = {}, acc2 = {}, acc3 = {};
  for (int k = 0; k < CH; k += 4){
    int kA = k + 2*hl;
    v2f a;
    a.x = fb[(size_t)kA*NPTS     + n0 + l15];
    a.y = fb[(size_t)(kA+1)*NPTS + n0 + l15];
    v2f b0, b1v, b2, b3;
    b0.x  = w1[kA*HCH + 0  + l15]; b0.y  = w1[(kA+1)*HCH + 0  + l15];
    b1v.x = w1[kA*HCH + 16 + l15]; b1v.y = w1[(kA+1)*HCH + 16 + l15];
    b2.x  = w1[kA*HCH + 32 + l15]; b2.y  = w1[(kA+1)*HCH + 32 + l15];
    b3.x  = w1[kA*HCH + 48 + l15]; b3.y  = w1[(kA+1)*HCH + 48 + l15];
    acc0 = __builtin_amdgcn_wmma_f32_16x16x4_f32(false, a, false, b0,  (short)0, acc0, false, false);
    acc1 = __builtin_amdgcn_wmma_f32_16x16x4_f32(false, a, false, b1v, (short)0, acc1, false, false);
    acc2 = __builtin_amdgcn_wmma_f32_16x16x4_f32(false, a, false, b2,  (short)0, acc2, false, false);
    acc3 = __builtin_amdgcn_wmma_f32_16x16x4_f32(false, a, false, b3,  (short)0, acc3, false, false);
  }
  #pragma unroll
  for (int i = 0; i < 8; ++i){
    int row = r0 + i + 8*hl;
    Y[(size_t)row*HCH + 0  + l15] = acc0[i];
    Y[(size_t)row*HCH + 16 + l15] = acc1[i];
    Y[(size_t)row*HCH + 32 + l15] = acc2[i];
    Y[(size_t)row*HCH + 48 + l15] = acc3[i];
  }
}

// ------------- per-column mean / rstd over 65536 rows ---------------------
__global__ void k_colstats(const float* __restrict__ Y, float* __restrict__ st){
  int j = blockIdx.x;                // 0..63
  float s = 0.f, ss = 0.f;
  for (int r = threadIdx.x; r < ROWS; r += blockDim.x){
    float v = Y[(size_t)r*HCH + j]; s += v; ss += v*v;
  }
  __shared__ float sh0[256], sh1[256];
  sh0[threadIdx.x] = s; sh1[threadIdx.x] = ss; __syncthreads();
  for (int off = 128; off > 0; off >>= 1){
    if ((int)threadIdx.x < off){ sh0[threadIdx.x]+=sh0[threadIdx.x+off]; sh1[threadIdx.x]+=sh1[threadIdx.x+off]; }
    __syncthreads();
  }
  if (threadIdx.x == 0){
    float mean = sh0[0] / (float)ROWS;
    float var  = sh1[0] / (float)ROWS - mean*mean;
    st[j*2+0] = mean;
    st[j*2+1] = rsqrtf(var + EPSF);
  }
}

// -------- h = gelu(bn(Y)); heatmap = h @ w2 (64x10), per-row --------------
__global__ void k_bn_gelu_gemm2(const float* __restrict__ Y, const float* __restrict__ st,
                                const float* __restrict__ g1, const float* __restrict__ b1,
                                const float* __restrict__ w2, float* __restrict__ heatmap){
  int r = blockIdx.x*blockDim.x + threadIdx.x;     // < 65536
  float h[HCH];
  #pragma unroll 8
  for (int j = 0; j < HCH; ++j){
    float v = Y[(size_t)r*HCH + j];
    h[j] = gelu_tanh((v - st[j*2]) * st[j*2+1] * g1[j] + b1[j]);
  }
  float out[NCLS];
  #pragma unroll
  for (int s = 0; s < NCLS; ++s) out[s] = 0.f;
  for (int j = 0; j < HCH; ++j){
    float hv = h[j];
    #pragma unroll
    for (int s = 0; s < NCLS; ++s) out[s] += hv * w2[j*NCLS + s];
  }
  #pragma unroll
  for (int s = 0; s < NCLS; ++s) heatmap[(size_t)r*NCLS + s] = out[s];
}

// -------- heatmap_preds[b,s] = mean_n heatmap[b,n,s] (deterministic) ------
__global__ void k_preds(const float* __restrict__ heatmap, float* __restrict__ preds){
  int bs = blockIdx.x;             // 0..39
  int b = bs / NCLS, s = bs % NCLS;
  float acc = 0.f;
  for (int n = threadIdx.x; n < NPTS; n += blockDim.x)
    acc += heatmap[((size_t)b*NPTS + n)*NCLS + s];
  __shared__ float sh[256];
  sh[threadIdx.x] = acc; __syncthreads();
  for (int off = 128; off > 0; off >>= 1){
    if ((int)threadIdx.x < off) sh[threadIdx.x] += sh[threadIdx.x+off];
    __syncthreads();
  }
  if (threadIdx.x == 0) preds[bs] = sh[0] * (1.0f/(float)NPTS);
}

__global__ void k_labels(const float* __restrict__ preds, int* __restrict__ labels){
  int b = threadIdx.x;
  if (b < BATCH){
    int best = 0; float bv = preds[b*NCLS];
    for (int s = 1; s < NCLS; ++s){
      float v = preds[b*NCLS + s];
      if (v > bv){ bv = v; best = s; }   // strict > == argmax first-index tie-break
    }
    labels[b] = best;
  }
}

// -------- exact stable top-256 per batch via rank counting in LDS ---------
__global__ __launch_bounds__(1024) void k_topk(const float* __restrict__ heatmap,
                                               const int* __restrict__ labels,
                                               int* __restrict__ qidx){
  int b = blockIdx.x;
  int lab = labels[b];
  __shared__ float vals[NPTS];     // 64 KB (320 KB/WGP available on CDNA5)
  for (int n = threadIdx.x; n < NPTS; n += blockDim.x)
    vals[n] = heatmap[((size_t)b*NPTS + n)*NCLS + lab];
  __syncthreads();
  for (int n = threadIdx.x; n < NPTS; n += blockDim.x){
    float v = vals[n];
    int rank = 0;
    for (int j = 0; j < NPTS; ++j){
      float u = vals[j];
      rank += (u > v) || (u == v && j < n);
    }
    if (rank < NQ) qidx[b*NQ + rank] = n;
  }
}

// -------- gather query points + query feats --------------------------------
__global__ void k_gather(const float* __restrict__ points, const float* __restrict__ feats,
                         const int* __restrict__ qidx,
                         float* __restrict__ query, float* __restrict__ qfeats){
  int bq = blockIdx.x;             // b*256 + q
  int b = bq >> 8, q = bq & 255;
  int idx = qidx[bq];
  int c = threadIdx.x;             // 0..255
  qfeats[((size_t)b*CH + c)*NQ + q] = feats[((size_t)b*CH + c)*NPTS + idx];
  if (c < 3) query[((size_t)b*3 + c)*NQ + q] = points[((size_t)b*3 + c)*NPTS + idx];
}

// ------------- GEMM-M: m[b,d,n] = sum_c wm[c,d]*feats[b,c,n] ---------------
__global__ void k_gemm_m(const float* __restrict__ feats, const float* __restrict__ wm,
                         float* __restrict__ m_out){
  int wid  = (blockIdx.x*blockDim.x + threadIdx.x) >> 5;
  int lane = threadIdx.x & 31;
  int hl   = lane >> 4;
  int l15  = lane & 15;
  int bb  = wid >> 12;             // 4096 waves per batch
  int rem = wid & 4095;
  int d0  = (rem >> 8) << 4;       // 16 d-tiles
  int n0  = (rem & 255) << 6;      // 256 n-strips of 64
  const float* fb = feats + (size_t)bb*CH*NPTS;
  float* mo = m_out + (size_t)bb*CH*NPTS;
  v8f acc0 = {}, acc1 = {}, acc2 = {}, acc3 = {};
  for (int k = 0; k < CH; k += 4){
    int kA = k + 2*hl;
    v2f a;
    a.x = wm[(size_t)kA*CH     + d0 + l15];   // A[d,c] = wm[c,d], L2-resident
    a.y = wm[(size_t)(kA+1)*CH + d0 + l15];
    const float* p0 = fb + (size_t)kA*NPTS     + n0 + l15;
    const float* p1 = fb + (size_t)(kA+1)*NPTS + n0 + l15;
    __builtin_prefetch(fb + (size_t)(kA+4)*NPTS + n0 + l15, 0, 1);
    v2f b0, b1v, b2, b3;
    b0.x  = p0[0];  b0.y  = p1[0];
    b1v.x = p0[16]; b1v.y = p1[16];
    b2.x  = p0[32]; b2.y  = p1[32];
    b3.x  = p0[48]; b3.y  = p1[48];
    acc0 = __builtin_amdgcn_wmma_f32_16x16x4_f32(false, a, false, b0,  (short)0, acc0, false, false);
    acc1 = __builtin_amdgcn_wmma_f32_16x16x4_f32(false, a, false, b1v, (short)0, acc1, false, false);
    acc2 = __builtin_amdgcn_wmma_f32_16x16x4_f32(false, a, false, b2,  (short)0, acc2, false, false);
    acc3 = __builtin_amdgcn_wmma_f32_16x16x4_f32(false, a, false, b3,  (short)0, acc3, false, false);
  }
  #pragma unroll
  for (int i = 0; i < 8; ++i){
    int d = d0 + i + 8*hl;
    mo[(size_t)d*NPTS + n0 + 0  + l15] = acc0[i];
    mo[(size_t)d*NPTS + n0 + 16 + l15] = acc1[i];
    mo[(size_t)d*NPTS + n0 + 32 + l15] = acc2[i];
    mo[(size_t)d*NPTS + n0 + 48 + l15] = acc3[i];
  }
}

// -------- per-channel mean/rstd over (B,N) for m ---------------------------
__global__ void k_mstats(const float* __restrict__ m_raw, float* __restrict__ st){
  int d = blockIdx.x;              // 0..255
  float s = 0.f, ss = 0.f;
  for (int i = threadIdx.x; i < BATCH*NPTS; i += blockDim.x){
    int b = i / NPTS, n = i - b*NPTS;
    float v = m_raw[((size_t)b*CH + d)*NPTS + n];
    s += v; ss += v*v;
  }
  __shared__ float sh0[256], sh1[256];
  sh0[threadIdx.x] = s; sh1[threadIdx.x] = ss; __syncthreads();
  for (int off = 128; off > 0; off >>= 1){
    if ((int)threadIdx.x < off){ sh0[threadIdx.x]+=sh0[threadIdx.x+off]; sh1[threadIdx.x]+=sh1[threadIdx.x+off]; }
    __syncthreads();
  }
  if (threadIdx.x == 0){
    float mean = sh0[0] / (float)(BATCH*NPTS);
    float var  = sh1[0] / (float)(BATCH*NPTS) - mean*mean;
    st[d*2+0] = mean;
    st[d*2+1] = rsqrtf(var + EPSF);
  }
}

// -------- in-place m = gelu(bn(m_raw)) ------------------------------------
__global__ void k_mfinal(float* __restrict__ m, const float* __restrict__ st,
                         const float* __restrict__ gm, const float* __restrict__ bm){
  size_t i = (size_t)blockIdx.x*blockDim.x + threadIdx.x;   // < B*CH*NPTS
  int d = (int)((i / NPTS) % CH);
  float x = m[i];
  m[i] = gelu_tanh((x - st[d*2]) * st[d*2+1] * gm[d] + bm[d]);
}

// -------- ball query, one wave per (b,q), both radii in one pass ----------
__global__ void k_ballquery(const float* __restrict__ points, const float* __restrict__ query,
                            float* __restrict__ g0, float* __restrict__ g1){
  int wid  = (blockIdx.x*blockDim.x + threadIdx.x) >> 5;    // < 1024
  int lane = threadIdx.x & 31;
  int b = wid >> 8, q = wid & 255;
  const float* px = points + (size_t)b*3*NPTS;
  float qx = query[((size_t)b*3 + 0)*NQ + q];
  float qy = query[((size_t)b*3 + 1)*NQ + q];
  float qz = query[((size_t)b*3 + 2)*NQ + q];
  float* o0 = g0 + (size_t)(b*NQ + q)*32;
  float* o1 = g1 + (size_t)(b*NQ + q)*64;
  int cnt0 = 0, cnt1 = 0, first0 = NPTS-1, first1 = NPTS-1;
  const float r0sq = 0.2f*0.2f, r1sq = 0.4f*0.4f;
  for (int base = 0; base < NPTS; base += 32){
    int n = base + lane;
    float dx = px[n] - qx;
    float dy = px[NPTS + n] - qy;
    float dz = px[2*NPTS + n] - qz;
    float d2 = dx*dx + dy*dy + dz*dz;
    bool in0 = (d2 <= r0sq), in1 = (d2 <= r1sq);
    unsigned m0 = (unsigned)__ballot(in0);
    unsigned m1 = (unsigned)__ballot(in1);
    unsigned lmask = (lane == 31) ? 0x7fffffffu : ((1u << lane) - 1u);
    if (m0 && cnt0 == 0) first0 = base + __ffs(m0) - 1;
    if (m1 && cnt1 == 0) first1 = base + __ffs(m1) - 1;
    int slot0 = cnt0 + __popc(m0 & lmask);
    int slot1 = cnt1 + __popc(m1 & lmask);
    if (in0 && slot0 < 32) o0[slot0] = (float)n;
    if (in1 && slot1 < 64) o1[slot1] = (float)n;
    cnt0 += __popc(m0);
    cnt1 += __popc(m1);
    if (cnt0 >= 32 && cnt1 >= 64) break;
  }
  // padding: repeat first in-radius index (or N-1 if none) in empty slots
  float pad0 = (float)first0, pad1 = (float)first1;
  for (int s = cnt0 + lane; s < 32; s += 32) o0[s] = pad0;
  for (int s = cnt1 + lane; s < 64; s += 32) o1[s] = pad1;
}

extern "C" void kernel_launch(void* const* d_in, const int* in_sizes, int n_in,
                              void* d_out, int out_size, void* d_ws, size_t ws_size,
                              hipStream_t stream){
  const float* points = (const float*)d_in[0];   // (4,3,16384)
  const float* feats  = (const float*)d_in[1];   // (4,256,16384)
  const float* w1     = (const float*)d_in[2];   // (256,64)
  const float* g1     = (const float*)d_in[3];   // (64)
  const float* b1     = (const float*)d_in[4];   // (64)
  const float* w2     = (const float*)d_in[5];   // (64,10)
  const float* wm     = (const float*)d_in[6];   // (256,256)
  const float* gm     = (const float*)d_in[7];   // (256)
  const float* bm     = (const float*)d_in[8];   // (256)

  float* out = (float*)d_out;
  // output layout (flat, return order)
  float* o_query = out;                               // 4*3*256      = 3072
  float* o_qf    = out + 3072;                        // 4*256*256    = 262144
  float* o_pred  = out + 265216;                      // 4*10         = 40
  float* o_m     = out + 265256;                      // 4*256*16384  = 16777216
  float* o_g0    = out + 17042472;                    // 4*256*32     = 32768
  float* o_g1    = out + 17075240;                    // 4*256*64     = 65536

  float* ws = (float*)d_ws;
  float* Y       = ws;                    // 65536*64  = 4194304
  float* heatmap = ws + 4194304;          // 65536*10  = 655360
  float* cst     = ws + 4849664;          // 64*2
  float* mst     = ws + 4849792;          // 256*2
  int*   labels  = (int*)(ws + 4850304);  // 4
  int*   qidx    = (int*)(ws + 4850308);  // 4*256

  // 1) Y = X @ w1 (WMMA f32)
  k_gemm1<<<512, 256, 0, stream>>>(feats, w1, Y);
  // 2) per-column BN stats
  k_colstats<<<HCH, 256, 0, stream>>>(Y, cst);
  // 3) h = gelu(bn(Y)); heatmap = h @ w2
  k_bn_gelu_gemm2<<<ROWS/256, 256, 0, stream>>>(Y, cst, g1, b1, w2, heatmap);
  // 4) heatmap_preds + labels
  k_preds<<<BATCH*NCLS, 256, 0, stream>>>(heatmap, o_pred);
  k_labels<<<1, 32, 0, stream>>>(o_pred, labels);
  // 5) exact stable top-256 per batch
  k_topk<<<BATCH, 1024, 0, stream>>>(heatmap, labels, qidx);
  // 6) gathers
  k_gather<<<BATCH*NQ, 256, 0, stream>>>(points, feats, qidx, o_query, o_qf);
  // 7) m_raw = wm^T @ feats (WMMA f32) directly into d_out m region
  k_gemm_m<<<2048, 256, 0, stream>>>(feats, wm, o_m);
  // 8) per-channel BN stats for m, then in-place BN+GELU
  k_mstats<<<CH, 256, 0, stream>>>(o_m, mst);
  k_mfinal<<<(BATCH*CH*NPTS)/256, 256, 0, stream>>>(o_m, mst, gm, bm);
  // 9) ball queries (reads query coords already written to d_out)
  k_ballquery<<<(BATCH*NQ*32)/256, 256, 0, stream>>>(points, o_query, o_g0, o_g1);
}